// ModelSAGESample_40097814676057
// MI455X (gfx1250) — compile-verified
//
#include <hip/hip_runtime.h>
#include <hip/hip_bf16.h>

typedef __attribute__((ext_vector_type(2))) float v2f;
typedef __attribute__((ext_vector_type(8))) float v8f;

#define N_SRC1 292864
#define N_DST1 11264
#define N_E1   281600
#define N_DST2 1024
#define N_E2   10240
#define DIM    256   // IN == H == 256
#define COUT   47

// ---------------------------------------------------------------------------
// Segment-mean aggregation. dst[] is sorted, so each block binary-searches the
// edge range belonging to its destination row: no atomics, deterministic, and
// each gathered source row is streamed exactly once (coalesced 1KB reads).
// grid = n_dst blocks, block = 256 threads (one per feature).
// ---------------------------------------------------------------------------
__global__ __launch_bounds__(256) void agg_mean_kernel(
    const float* __restrict__ feat, const int* __restrict__ src,
    const int* __restrict__ dst, int n_edges, float* __restrict__ out)
{
    const int row = blockIdx.x;
    const int col = threadIdx.x;

    // lower_bound(dst, row)
    int lo = 0, hi = n_edges;
    while (lo < hi) { int mid = (lo + hi) >> 1; if (dst[mid] <  row) lo = mid + 1; else hi = mid; }
    const int e0 = lo;
    // upper_bound(dst, row)
    hi = n_edges;
    while (lo < hi) { int mid = (lo + hi) >> 1; if (dst[mid] <= row) lo = mid + 1; else hi = mid; }
    const int e1 = lo;

    float acc = 0.0f;
    for (int e = e0; e < e1; ++e) {
        const int s = src[e];
        acc += feat[(size_t)s * DIM + col];
    }
    const int deg = e1 - e0;
    out[(size_t)row * DIM + col] = (deg > 0) ? acc * (1.0f / (float)deg) : 0.0f;
}

// ---------------------------------------------------------------------------
// Layer-1 fused GEMM: h1 = relu(A0 @ W0 + A1 @ W1 + bias), all 256-wide, f32.
// Uses V_WMMA_F32_16X16X4_F32 (full f32 precision on the matrix pipe).
// Block = 256 threads = 8 waves -> 64 rows x 256 cols per block.
// Wave tile = 16 rows x 128 cols = 8 accumulator fragments.
// W staged through LDS in 32x256 chunks (lanes read consecutive cols: no bank
// conflicts). A fragments are float2 global loads per the 16x4 f32 A layout:
//   VGPR0 = K {0 | 2}, VGPR1 = K {1 | 3} across lane halves.
// ---------------------------------------------------------------------------
__global__ __launch_bounds__(256) void gemm1_kernel(
    const float* __restrict__ A0, const float* __restrict__ A1,
    const float* __restrict__ W0, const float* __restrict__ W1,
    const float* __restrict__ bias, float* __restrict__ out)
{
    __shared__ float lB[32][DIM];

    const int tid      = threadIdx.x;
    const int wave     = tid >> 5;
    const int lane     = tid & 31;
    const int half     = lane >> 4;   // 0: lanes 0-15, 1: lanes 16-31
    const int r        = lane & 15;
    const int row_wave = wave >> 1;   // 0..3
    const int col_wave = wave & 1;    // 0..1
    const int rowbase  = blockIdx.x * 64 + row_wave * 16;
    const int colbase  = col_wave * 128;

    v8f c[8];
    #pragma unroll
    for (int i = 0; i < 8; ++i)
        c[i] = (v8f){0.f, 0.f, 0.f, 0.f, 0.f, 0.f, 0.f, 0.f};

    for (int phase = 0; phase < 2; ++phase) {
        const float* __restrict__ A = phase ? A1 : A0;
        const float* __restrict__ W = phase ? W1 : W0;
        const float* __restrict__ arow = A + (size_t)(rowbase + r) * DIM;

        for (int kc = 0; kc < DIM; kc += 32) {
            __syncthreads();
            // Cooperative load of W[kc:kc+32][0:256] into LDS (float4, coalesced).
            #pragma unroll
            for (int it = 0; it < 8; ++it) {
                const int idx = tid + it * 256;       // 0..2047 float4 slots
                const int rw  = idx >> 6;             // 64 float4 per row
                const int c4  = (idx & 63) << 2;
                const float4 v = *(const float4*)(W + (size_t)(kc + rw) * DIM + c4);
                *(float4*)&lB[rw][c4] = v;
            }
            __syncthreads();

            #pragma unroll
            for (int kk = 0; kk < 32; kk += 4) {
                const int kpos = kk + half * 2;
                v2f a;
                a.x = arow[kc + kpos];
                a.y = arow[kc + kpos + 1];
                #pragma unroll
                for (int nt = 0; nt < 8; ++nt) {
                    const int col = colbase + nt * 16 + r;
                    v2f b;
                    b.x = lB[kpos][col];
                    b.y = lB[kpos + 1][col];
                    c[nt] = __builtin_amdgcn_wmma_f32_16x16x4_f32(
                        false, a, false, b, (short)0, c[nt], false, false);
                }
            }
        }
    }

    // Epilogue: bias + ReLU. C/D layout: VGPR i holds M=i (lanes 0-15) and
    // M=i+8 (lanes 16-31); N = lane within half.
    #pragma unroll
    for (int nt = 0; nt < 8; ++nt) {
        const int col = colbase + nt * 16 + r;
        const float bv = bias[col];
        #pragma unroll
        for (int i = 0; i < 8; ++i) {
            const int row = rowbase + i + 8 * half;
            float v = c[nt][i] + bv;
            out[(size_t)row * DIM + col] = v > 0.f ? v : 0.f;
        }
    }
}

// ---------------------------------------------------------------------------
// Layer-2 GEMM: out[1024 x 47] = A0 @ W0 + A1 @ W1 + bias (no ReLU).
// 16 blocks x 4 waves; each wave owns a 16-row strip and 3 column tiles
// (48 cols, masked to 47). W is tiny and hot in cache: direct global B loads.
// ---------------------------------------------------------------------------
__global__ __launch_bounds__(128) void gemm2_kernel(
    const float* __restrict__ A0, const float* __restrict__ A1,
    const float* __restrict__ W0, const float* __restrict__ W1,
    const float* __restrict__ bias, float* __restrict__ out)
{
    const int tid     = threadIdx.x;
    const int wave    = tid >> 5;
    const int lane    = tid & 31;
    const int half    = lane >> 4;
    const int r       = lane & 15;
    const int rowbase = (blockIdx.x * 4 + wave) * 16;

    v8f c[3];
    #pragma unroll
    for (int i = 0; i < 3; ++i)
        c[i] = (v8f){0.f, 0.f, 0.f, 0.f, 0.f, 0.f, 0.f, 0.f};

    for (int phase = 0; phase < 2; ++phase) {
        const float* __restrict__ A = phase ? A1 : A0;
        const float* __restrict__ W = phase ? W1 : W0;
        const float* __restrict__ arow = A + (size_t)(rowbase + r) * DIM;

        for (int k = 0; k < DIM; k += 4) {
            const int kpos = k + half * 2;
            v2f a;
            a.x = arow[kpos];
            a.y = arow[kpos + 1];
            #pragma unroll
            for (int nt = 0; nt < 3; ++nt) {
                const int col = nt * 16 + r;
                v2f b;
                b.x = (col < COUT) ? W[(size_t)kpos * COUT + col] : 0.f;
                b.y = (col < COUT) ? W[(size_t)(kpos + 1) * COUT + col] : 0.f;
                c[nt] = __builtin_amdgcn_wmma_f32_16x16x4_f32(
                    false, a, false, b, (short)0, c[nt], false, false);
            }
        }
    }

    #pragma unroll
    for (int nt = 0; nt < 3; ++nt) {
        const int col = nt * 16 + r;
        if (col < COUT) {
            const float bv = bias[col];
            #pragma unroll
            for (int i = 0; i < 8; ++i) {
                const int row = rowbase + i + 8 * half;
                out[(size_t)row * COUT + col] = c[nt][i] + bv;
            }
        }
    }
}

// ---------------------------------------------------------------------------
// Inputs (setup_inputs order):
//  0:x  1:src1  2:dst1  3:src2  4:dst2  5:W_self1  6:W_neigh1  7:b1
//  8:W_self2  9:W_neigh2  10:b2
// ---------------------------------------------------------------------------
extern "C" void kernel_launch(void* const* d_in, const int* in_sizes, int n_in,
                              void* d_out, int out_size, void* d_ws, size_t ws_size,
                              hipStream_t stream)
{
    const float* x    = (const float*)d_in[0];
    const int*   src1 = (const int*)  d_in[1];
    const int*   dst1 = (const int*)  d_in[2];
    const int*   src2 = (const int*)  d_in[3];
    const int*   dst2 = (const int*)  d_in[4];
    const float* Ws1  = (const float*)d_in[5];
    const float* Wn1  = (const float*)d_in[6];
    const float* b1   = (const float*)d_in[7];
    const float* Ws2  = (const float*)d_in[8];
    const float* Wn2  = (const float*)d_in[9];
    const float* b2   = (const float*)d_in[10];
    float*       out  = (float*)d_out;

    float* ws  = (float*)d_ws;
    float* hn1 = ws;                                   // 11264 x 256
    float* h1  = ws + (size_t)N_DST1 * DIM;            // 11264 x 256
    float* hn2 = ws + (size_t)2 * N_DST1 * DIM;        // 1024 x 256

    // Layer 1
    agg_mean_kernel<<<N_DST1, 256, 0, stream>>>(x, src1, dst1, N_E1, hn1);
    gemm1_kernel<<<N_DST1 / 64, 256, 0, stream>>>(x, hn1, Ws1, Wn1, b1, h1);

    // Layer 2
    agg_mean_kernel<<<N_DST2, 256, 0, stream>>>(h1, src2, dst2, N_E2, hn2);
    gemm2_kernel<<<N_DST2 / 64, 128, 0, stream>>>(h1, hn2, Ws2, Wn2, b2, out);
}